// EqualizedModulatedConv2d_23021024707464
// MI455X (gfx1250) — compile-verified
//
#include <hip/hip_runtime.h>
#include <math.h>

typedef __attribute__((ext_vector_type(2))) float v2f;
typedef __attribute__((ext_vector_type(8))) float v8f;

#define IC 512
#define OC 512
#define NB 16
#define HH 64
#define WW 64

// ---- LDS address helper: generic pointer -> 32-bit LDS byte offset --------
typedef __attribute__((address_space(3))) const void lds_cvoid;
__device__ __forceinline__ unsigned lds_addr(const void* p) {
  return (unsigned)(unsigned long long)(lds_cvoid*)p;
}

// ---- async 4B copy global -> LDS (GVS mode), tracked by ASYNCcnt ----------
__device__ __forceinline__ void async_copy_b32(unsigned lds_off,
                                               const float* sbase,
                                               int vbyte_off) {
  asm volatile("global_load_async_to_lds_b32 %0, %1, %2"
               :
               : "v"(lds_off), "v"(vbyte_off), "s"(sbase)
               : "memory");
}
__device__ __forceinline__ void wait_asynccnt0() {
  asm volatile("s_wait_asynccnt 0x0" ::: "memory");
}

// ---------------------------------------------------------------------------
// t[b,i] = elr * ( lin_scale * dot(style[b,:], fc_weight[i,:]) + fc_bias[i] )
// ---------------------------------------------------------------------------
__global__ __launch_bounds__(256) void style_kernel(
    const float* __restrict__ style, const float* __restrict__ fcw,
    const float* __restrict__ fcb, float* __restrict__ t) {
  int idx = blockIdx.x * 256 + threadIdx.x;           // 16*512 = 8192
  if (idx >= NB * IC) return;
  int b = idx >> 9;
  int i = idx & 511;
  const float lin_scale = 0.0625f;                    // sqrt(2/512)
  const float elr = 0.020833333333333332f;            // sqrt(2/(512*9)) = 1/48
  float acc = 0.f;
  const float* sp = style + b * 512;
  const float* wp = fcw + (size_t)i * 512;
  #pragma unroll 4
  for (int k = 0; k < 512; ++k) acc += sp[k] * wp[k];
  t[idx] = elr * (acc * lin_scale + fcb[i]);
}

// ---------------------------------------------------------------------------
// w2[o,i] = sum_{kh,kw} W[o,i,kh,kw]^2
// ---------------------------------------------------------------------------
__global__ __launch_bounds__(256) void w2_kernel(
    const float* __restrict__ weight, float* __restrict__ w2) {
  int idx = blockIdx.x * 256 + threadIdx.x;           // 512*512
  if (idx >= OC * IC) return;
  const float* wp = weight + (size_t)idx * 9;
  float acc = 0.f;
  #pragma unroll
  for (int k = 0; k < 9; ++k) acc += wp[k] * wp[k];
  w2[idx] = acc;
}

// ---------------------------------------------------------------------------
// norm[b,o] = rsqrt( sum_i w2[o,i] * t[b,i]^2 + 1e-8 )
// ---------------------------------------------------------------------------
__global__ __launch_bounds__(256) void norm_kernel(
    const float* __restrict__ w2, const float* __restrict__ t,
    float* __restrict__ nrm) {
  int idx = blockIdx.x * 256 + threadIdx.x;           // 16*512
  if (idx >= NB * OC) return;
  int b = idx >> 9;
  int o = idx & 511;
  const float* wp = w2 + (size_t)o * IC;
  const float* tp = t + b * IC;
  float acc = 0.f;
  #pragma unroll 4
  for (int i = 0; i < IC; ++i) {
    float ti = tp[i];
    acc += wp[i] * ti * ti;
  }
  nrm[idx] = rsqrtf(acc + 1e-8f);
}

// ---------------------------------------------------------------------------
// Implicit-GEMM 3x3 conv, fp32 WMMA 16x16x4.
//  - modulation t[b,i] folded into the A (weight) tile at staging time
//  - x rows copied raw to LDS with global_load_async_to_lds_b32 (ASYNCcnt)
//  - both LDS tiles pair-interleaved along K so fragments are ds_load_b64
// grid.x = B*H (one output row / WG), grid.y = OC/64; 256 thr = 8 waves.
// ---------------------------------------------------------------------------
__global__ __launch_bounds__(256) void conv_kernel(
    const float* __restrict__ x, const float* __restrict__ weight,
    const float* __restrict__ t, const float* __restrict__ nrm,
    float* __restrict__ out) {
  // xsB[kh(3)][pair(16)][col(66)][e(2)]  col 0 & 65 = halo (stay zero)
  __shared__ float xsB[3 * 16 * 66 * 2];              // 25344 B
  // wlA[kw(3)][pair(16)][o(64)][e(2)]   for current kh, scaled by t[b,i]
  __shared__ float wlA[3 * 16 * 64 * 2];              // 24576 B

  const int bh = blockIdx.x;
  const int b = bh >> 6;                              // batch
  const int h = bh & 63;                              // output row
  const int o0 = blockIdx.y * 64;                     // output-channel base

  const int tid = threadIdx.x;
  const int wave = tid >> 5;
  const int lane = tid & 31;
  const int l15 = lane & 15;
  const int lhi = lane >> 4;                          // 0/1
  const int m_sub = wave & 3;                         // 16-row M subtile
  const int n_sub0 = (wave >> 2) * 2;                 // first 16-col N subtile

  v8f acc0 = {};
  v8f acc1 = {};

  const float* tb = t + b * IC;
  const unsigned xs_base = lds_addr(xsB);

  // zero x tile once: halo cols + OOB kh slabs stay zero forever
  for (int e = tid; e < 3 * 16 * 66 * 2; e += 256) xsB[e] = 0.f;

  for (int ic0 = 0; ic0 < IC; ic0 += 32) {
    __syncthreads();  // prev block's compute done (covers initial zero too)

    // ---- async-copy raw x rows into interleaved LDS tile ----
    #pragma unroll
    for (int kh = 0; kh < 3; ++kh) {
      int r = h + kh - 1;
      if ((unsigned)r < 64u) {                        // uniform per block
        for (int e = tid; e < 32 * 64; e += 256) {
          int i = e >> 6;                             // 0..31
          int c = e & 63;                             // 0..63
          int go = (((b * IC + ic0 + i) * 64 + r) * 64 + c) * 4;
          unsigned lo =
              xs_base +
              ((((kh * 16 + (i >> 1)) * 66 + (c + 1)) << 1) + (i & 1)) * 4;
          async_copy_b32(lo, x, go);
        }
      }
    }

    for (int kh = 0; kh < 3; ++kh) {
      if (kh) __syncthreads();                        // prev kh done with wlA
      // ---- stage weight slice (x t[b,i]) : wlA[kw][i/2][o][i&1] ----
      for (int p = tid; p < 64 * 32; p += 256) {
        int o_local = p & 63;
        int i = p >> 6;
        float ti = tb[ic0 + i];
        const float* wp =
            weight + (((size_t)(o0 + o_local) * IC + (ic0 + i)) * 9 + kh * 3);
        int base = ((i >> 1) * 64 + o_local) * 2 + (i & 1);
        wlA[(0 * 16 * 64) * 2 + base] = wp[0] * ti;
        wlA[(1 * 16 * 64) * 2 + base] = wp[1] * ti;
        wlA[(2 * 16 * 64) * 2 + base] = wp[2] * ti;
      }
      if (kh == 0) wait_asynccnt0();                  // x tile landed
      __syncthreads();                                // staging visible
      const float* xrow = &xsB[kh * (16 * 66 * 2)];
      for (int kw = 0; kw < 3; ++kw) {
        #pragma unroll
        for (int kk = 0; kk < 32; kk += 4) {
          const int pair = (kk >> 1) + lhi;           // K pair for this lane
          // A frag: A[m=l15][k=2*lhi+e] = (W*t)[o0+m_sub*16+m][ic0+2*pair+e]
          v2f a = *(const v2f*)&wlA[((kw * 16 + pair) * 64 + m_sub * 16 + l15)
                                    << 1];
          // B frag: B[k=2*lhi+e][n=l15] = x[ic0+2*pair+e][n_glob + kw - 1]
          const float* xr = xrow + ((pair * 66 + kw + l15) << 1);
          v2f b0 = *(const v2f*)&xr[(n_sub0 * 16) << 1];
          v2f b1 = *(const v2f*)&xr[((n_sub0 + 1) * 16) << 1];
          acc0 = __builtin_amdgcn_wmma_f32_16x16x4_f32(
              false, a, false, b0, (short)0, acc0, false, false);
          acc1 = __builtin_amdgcn_wmma_f32_16x16x4_f32(
              false, a, false, b1, (short)0, acc1, false, false);
        }
      }
    }
  }

  // ---- epilogue: demodulate and store ----
  // D layout: vgpr r -> M = r + 8*lhi, N = l15
  #pragma unroll
  for (int r = 0; r < 8; ++r) {
    int m = r + 8 * lhi;
    int o = o0 + m_sub * 16 + m;
    float nz = nrm[b * OC + o];
    size_t base = (((size_t)b * OC + o) * 64 + h) * 64;
    int n0 = n_sub0 * 16 + l15;
    out[base + n0] = acc0[r] * nz;
    out[base + n0 + 16] = acc1[r] * nz;
  }
}

extern "C" void kernel_launch(void* const* d_in, const int* in_sizes, int n_in,
                              void* d_out, int out_size, void* d_ws,
                              size_t ws_size, hipStream_t stream) {
  const float* x = (const float*)d_in[0];       // [16,512,64,64]
  const float* style = (const float*)d_in[1];   // [16,512]
  const float* weight = (const float*)d_in[2];  // [512,512,3,3]
  const float* fcw = (const float*)d_in[3];     // [512,512]
  const float* fcb = (const float*)d_in[4];     // [512]
  float* out = (float*)d_out;                   // [16,512,64,64]

  float* ws = (float*)d_ws;
  float* t = ws;                                // 16*512
  float* w2 = ws + NB * IC;                     // 512*512
  float* nrm = w2 + (size_t)OC * IC;            // 16*512

  style_kernel<<<(NB * IC + 255) / 256, 256, 0, stream>>>(style, fcw, fcb, t);
  w2_kernel<<<(OC * IC + 255) / 256, 256, 0, stream>>>(weight, w2);
  norm_kernel<<<(NB * OC + 255) / 256, 256, 0, stream>>>(w2, t, nrm);

  dim3 grid(NB * HH, OC / 64);
  conv_kernel<<<grid, 256, 0, stream>>>(x, weight, t, nrm, out);
}